// SGCBlock_1365799600619
// MI455X (gfx1250) — compile-verified
//
#include <hip/hip_runtime.h>
#include <hip/hip_bf16.h>
#include <math.h>

typedef float v2f __attribute__((ext_vector_type(2)));
typedef float v8f __attribute__((ext_vector_type(8)));

#define BB  32
#define CC  128
#define NN  1024
#define KNN 9

// ---------------------------------------------------------------------------
// Kernel A: per-token squared norms  sq[b,n] = sum_c x[b,c,n]^2
// x layout (B, C, N); lanes stride over n -> coalesced.
// ---------------------------------------------------------------------------
__global__ void __launch_bounds__(256)
sq_kernel(const float* __restrict__ x, float* __restrict__ sq) {
    int gid = blockIdx.x * 256 + threadIdx.x;   // 0 .. B*N-1
    int b = gid >> 10;
    int n = gid & (NN - 1);
    const float* xb = x + (size_t)b * CC * NN + n;
    float s = 0.0f;
#pragma unroll 8
    for (int c = 0; c < CC; ++c) {
        float v = xb[(size_t)c * NN];
        s += v * v;
    }
    sq[gid] = s;
}

// ---------------------------------------------------------------------------
// Kernel B: fused Gram (fp32 WMMA) + distance + streaming top-9 + neighbor mean.
// One wave (32 threads) per (batch, 16-token row tile). Streams 64 column
// tiles; each tile = 32 x v_wmma_f32_16x16x4_f32 (K = 128).
// ---------------------------------------------------------------------------
__global__ void __launch_bounds__(32)
knn_agg_kernel(const float* __restrict__ x, const float* __restrict__ sq,
               float* __restrict__ agg) {
    __shared__ float tile[16][16];
    __shared__ float listd[16][2][KNN];
    __shared__ int   listi[16][2][KNN];
    __shared__ int   knn[16][KNN];

    const int b    = blockIdx.x >> 6;          // / (N/16)
    const int rb   = (blockIdx.x & 63) << 4;   // row tile base
    const int lane = threadIdx.x;
    const int half = lane >> 4;                // 0 | 1
    const int lr   = lane & 15;

    const float* xb  = x  + (size_t)b * CC * NN;
    const float* sqb = sq + b * NN;

    // Preload A fragments once: A[m=lr, k], k = 4*s + 2*half + {0,1}
    v2f afr[32];
#pragma unroll
    for (int s = 0; s < 32; ++s) {
        int k0 = 4 * s + 2 * half;
        afr[s].x = xb[(size_t)k0 * NN + rb + lr];
        afr[s].y = xb[(size_t)(k0 + 1) * NN + rb + lr];
    }

    // Row norms for this lane's 8 output rows in the C/D layout
    float sqr[8];
#pragma unroll
    for (int v = 0; v < 8; ++v) sqr[v] = sqb[rb + v + 8 * half];

    // Streaming top-9 state: lane covers row = lane>>1, 8 columns per tile
    float bd[KNN]; int bi[KNN];
#pragma unroll
    for (int j = 0; j < KNN; ++j) { bd[j] = 3.4e38f; bi[j] = 0; }
    const int trow  = lane >> 1;
    const int tcol0 = (lane & 1) * 8;

    for (int ct = 0; ct < 64; ++ct) {
        const int cb = ct << 4;
        v8f acc = {};
#pragma unroll
        for (int s = 0; s < 32; ++s) {
            int k0 = 4 * s + 2 * half;
            v2f bfr;
            bfr.x = xb[(size_t)k0 * NN + cb + lr];
            bfr.y = xb[(size_t)(k0 + 1) * NN + cb + lr];
            acc = __builtin_amdgcn_wmma_f32_16x16x4_f32(
                false, afr[s], false, bfr, (short)0, acc, false, false);
        }
        // d2 = sq[row] + sq[col] - 2*G ; stage distance tile through LDS
        float sqc = sqb[cb + lr];
#pragma unroll
        for (int v = 0; v < 8; ++v) {
            float d2 = sqr[v] + sqc - 2.0f * acc[v];
            tile[v + 8 * half][lr] = sqrtf(fmaxf(d2, 0.0f));
        }
        __syncthreads();
        // streaming sorted-insertion top-9 (half a row per lane)
#pragma unroll
        for (int j = 0; j < 8; ++j) {
            float v = tile[trow][tcol0 + j];
            int   m = cb + tcol0 + j;
            if (v < bd[KNN - 1]) {
                bd[KNN - 1] = v; bi[KNN - 1] = m;
#pragma unroll
                for (int q = KNN - 1; q > 0; --q) {
                    if (bd[q] < bd[q - 1]) {
                        float td = bd[q]; bd[q] = bd[q - 1]; bd[q - 1] = td;
                        int   ti = bi[q]; bi[q] = bi[q - 1]; bi[q - 1] = ti;
                    }
                }
            }
        }
        __syncthreads();
    }

    // Dump per-lane half-lists
#pragma unroll
    for (int j = 0; j < KNN; ++j) {
        listd[trow][lane & 1][j] = bd[j];
        listi[trow][lane & 1][j] = bi[j];
    }
    __syncthreads();

    // Merge the two sorted 9-lists per row (lanes 0..15); tie-break on index
    if (lane < 16) {
        int ia = 0, ib = 0;
#pragma unroll
        for (int t = 0; t < KNN; ++t) {
            float da  = listd[lane][0][ia], db2 = listd[lane][1][ib];
            int   iaI = listi[lane][0][ia], ibI = listi[lane][1][ib];
            bool takeA = (da < db2) || (da == db2 && iaI < ibI);
            knn[lane][t] = takeA ? iaI : ibI;
            if (takeA) ++ia; else ++ib;
        }
    }
    __syncthreads();

    // Neighbor mean: agg[b, rb+row, c] = mean_j x[b, c, knn[row][j]]
    // Wave-uniform row per iteration -> knn[] reads are scalar-friendly.
    float* aggb = agg + (size_t)b * NN * CC;
    for (int t = 0; t < 64; ++t) {
        int row = t >> 2;
        int c   = ((t & 3) << 5) + lane;
        float s = 0.0f;
#pragma unroll
        for (int j = 0; j < KNN; ++j)
            s += xb[(size_t)c * NN + knn[row][j]];
        aggb[(size_t)(rb + row) * CC + c] = s * (1.0f / 9.0f);
    }
}

// ---------------------------------------------------------------------------
// Kernel C: out[b,d,n] = sum_c agg[b,n,c] * w[d,c] + bias[d]   (fp32 WMMA)
// One wave per 16x16 output tile, K=128 in 32 steps; writes (B,C,N) layout.
// ---------------------------------------------------------------------------
__global__ void __launch_bounds__(32)
linear_kernel(const float* __restrict__ agg, const float* __restrict__ w,
              const float* __restrict__ bias, float* __restrict__ out) {
    const int bx   = blockIdx.x;
    const int b    = bx >> 9;           // / (64*8)
    const int r    = bx & 511;
    const int nb   = (r >> 3) << 4;     // token tile base
    const int db   = (r & 7) << 4;      // output-channel tile base
    const int lane = threadIdx.x;
    const int half = lane >> 4;
    const int lr   = lane & 15;

    const float* arow = agg + (size_t)b * NN * CC + (size_t)(nb + lr) * CC;
    const float* wrow = w + (size_t)(db + lr) * CC;

    v8f acc = {};
#pragma unroll
    for (int s = 0; s < 32; ++s) {
        int k0 = 4 * s + 2 * half;
        v2f a  = *(const v2f*)(arow + k0);   // contiguous float2 in c
        v2f bf = *(const v2f*)(wrow + k0);   // contiguous float2 in c
        acc = __builtin_amdgcn_wmma_f32_16x16x4_f32(
            false, a, false, bf, (short)0, acc, false, false);
    }
    float bv = bias[db + lr];
    float* ob = out + (size_t)b * CC * NN + (size_t)(db + lr) * NN + nb;
#pragma unroll
    for (int v = 0; v < 8; ++v) ob[v + 8 * half] = acc[v] + bv;
}

// ---------------------------------------------------------------------------
// Kernel D: per-channel BN stats over (B, N): mean[d], rstd[d]
// ---------------------------------------------------------------------------
__global__ void __launch_bounds__(256)
bnstats_kernel(const float* __restrict__ out, float* __restrict__ mean,
               float* __restrict__ rstd) {
    __shared__ float ssum[256], ssq[256];
    const int d   = blockIdx.x;
    const int tid = threadIdx.x;
    float s = 0.0f, q = 0.0f;
    for (int i = tid; i < BB * NN; i += 256) {
        int b = i >> 10, n = i & (NN - 1);
        float v = out[(size_t)b * CC * NN + (size_t)d * NN + n];
        s += v; q += v * v;
    }
    ssum[tid] = s; ssq[tid] = q;
    __syncthreads();
    for (int off = 128; off > 0; off >>= 1) {
        if (tid < off) { ssum[tid] += ssum[tid + off]; ssq[tid] += ssq[tid + off]; }
        __syncthreads();
    }
    if (tid == 0) {
        float m   = ssum[0] * (1.0f / (BB * NN));
        float var = ssq[0] * (1.0f / (BB * NN)) - m * m;
        mean[d] = m;
        rstd[d] = rsqrtf(var + 1e-5f);
    }
}

// ---------------------------------------------------------------------------
// Kernel E: y = relu(gamma*(out-mean)*rstd + beta + x), in-place on d_out
// ---------------------------------------------------------------------------
__global__ void __launch_bounds__(256)
bnrelu_kernel(const float* __restrict__ x, const float* __restrict__ gamma,
              const float* __restrict__ beta, const float* __restrict__ mean,
              const float* __restrict__ rstd, float* __restrict__ out) {
    int i = blockIdx.x * 256 + threadIdx.x;
    int d = (i >> 10) & (CC - 1);
    float v = out[i];
    float r = gamma[d] * (v - mean[d]) * rstd[d] + beta[d] + x[i];
    out[i] = fmaxf(r, 0.0f);
}

// ---------------------------------------------------------------------------
extern "C" void kernel_launch(void* const* d_in, const int* in_sizes, int n_in,
                              void* d_out, int out_size, void* d_ws, size_t ws_size,
                              hipStream_t stream) {
    const float* x     = (const float*)d_in[0];
    const float* w     = (const float*)d_in[1];
    const float* bias  = (const float*)d_in[2];
    const float* gamma = (const float*)d_in[3];
    const float* beta  = (const float*)d_in[4];
    // d_in[5] is k == 9 (compile-time constant here)

    float* out = (float*)d_out;
    float* ws  = (float*)d_ws;
    float* sq   = ws;                          // B*N           = 32768 floats
    float* agg  = ws + BB * NN;                // B*N*C         = 4194304 floats
    float* mean = agg + (size_t)BB * NN * CC;  // C             = 128 floats
    float* rstd = mean + CC;                   // C             = 128 floats

    sq_kernel     <<<(BB * NN) / 256, 256, 0, stream>>>(x, sq);
    knn_agg_kernel<<<BB * (NN / 16), 32, 0, stream>>>(x, sq, agg);
    linear_kernel <<<BB * (NN / 16) * (CC / 16), 32, 0, stream>>>(agg, w, bias, out);
    bnstats_kernel<<<CC, 256, 0, stream>>>(out, mean, rstd);
    bnrelu_kernel <<<(BB * CC * NN) / 256, 256, 0, stream>>>(x, gamma, beta, mean, rstd, out);
}